// RankingLoss_44135083934201
// MI455X (gfx1250) — compile-verified
//
#include <hip/hip_runtime.h>

// RankNet pairwise ranking loss, sort-free reformulation:
//   loss = sum over ordered (a,b) with lb[a]-lb[b] > TOL of log_sigmoid(ls[a]-ls[b])
// 16x16 tiles of the rank-2 difference matrices via V_WMMA_F32_16X16X4_F32.
// LDS staging via TDM (tensor_load_to_lds) when the builtin exists, else via
// async DMA (global_load_async_to_lds_b128 + s_wait_asynccnt).

typedef __attribute__((ext_vector_type(2))) float v2f;
typedef __attribute__((ext_vector_type(8))) float v8f;
typedef __attribute__((ext_vector_type(4))) unsigned int v4u;
typedef __attribute__((ext_vector_type(4))) int v4i;
typedef __attribute__((ext_vector_type(8))) int v8i;

#define TOLERANCE 0.01f
#define MAXN 8192
#define BLOCKS 512
#define THREADS 256

#if __has_builtin(__builtin_amdgcn_tensor_load_to_lds)
#define HAVE_TDM_BUILTIN 1
#else
#define HAVE_TDM_BUILTIN 0
#endif

__device__ __forceinline__ float fast_exp2(float x) {
#if __has_builtin(__builtin_amdgcn_exp2f)
  return __builtin_amdgcn_exp2f(x);   // raw v_exp_f32
#else
  return __builtin_exp2f(x);
#endif
}
__device__ __forceinline__ float fast_log2(float x) {
#if __has_builtin(__builtin_amdgcn_logf)
  return __builtin_amdgcn_logf(x);    // raw v_log_f32
#else
  return __builtin_log2f(x);
#endif
}

#if HAVE_TDM_BUILTIN
// 1D contiguous copy of n fp32 elements, global -> LDS, via one TDM descriptor.
// D# layout per CDNA5 ISA §8.3 (group0) / §8.4 (group1).
__device__ __forceinline__ void tdm_load_1d(const float* gsrc, void* ldst, int n) {
  unsigned long long ga = (unsigned long long)(uintptr_t)gsrc;
  unsigned int la = (unsigned int)(uintptr_t)ldst;  // low 32b of LDS aperture addr = LDS offset
  unsigned int dim0 = (unsigned int)n;
  v4u g0 = {
      1u,                                                  // count=1 (valid user D#)
      la,                                                  // lds_addr [63:32]
      (unsigned int)(ga & 0xffffffffu),                    // global_addr [95:64]
      (unsigned int)(((ga >> 32) & 0x01ffffffu) | (2u << 30))  // addr[56:32] | type=2
  };
  v8i g1 = {
      (int)(2u << 16),                    // workgroup_mask=0, data_size=2 (4B)
      (int)((dim0 & 0xffffu) << 16),      // atomic_barrier=0 | tensor_dim0[15:0]
      (int)((dim0 >> 16) | (1u << 16)),   // tensor_dim0[31:16] | tensor_dim1[15:0]=1
      (int)((dim0 & 0xffffu) << 16),      // tensor_dim1[31:16]=0 | tile_dim0=n
      1,                                  // tile_dim1=1 | tile_dim2=0
      (int)dim0,                          // tensor_dim0_stride[31:0]
      0, 0                                // stride0[47:32]=0 | stride1=0
  };
  v4i gz = {0, 0, 0, 0};
#if defined(__clang_major__) && (__clang_major__ >= 23)
  v8i gz8 = {0, 0, 0, 0, 0, 0, 0, 0};
  __builtin_amdgcn_tensor_load_to_lds(g0, g1, gz, gz, gz8, 0);  // 6-arg (clang-23 lane)
#else
  __builtin_amdgcn_tensor_load_to_lds(g0, g1, gz, gz, 0);       // 5-arg (ROCm 7.2)
#endif
}
#endif

__global__ __launch_bounds__(THREADS) void RankingLoss_pairs_kernel(
    const float* __restrict__ logits, const float* __restrict__ labels,
    float* __restrict__ partial, int n) {
  // 64 KB LDS: both arrays fully resident per workgroup (WGP has 320 KB).
  __shared__ float4 s_logit4[MAXN / 4];
  __shared__ float4 s_label4[MAXN / 4];
  __shared__ float s_wavesum[THREADS / 32];
  const float* s_logit = (const float*)s_logit4;
  const float* s_label = (const float*)s_label4;

  const int tid = threadIdx.x;
  const int lane = tid & 31;
  const int wave = tid >> 5;

  // gfx1250 prefetch path (global_prefetch_b8) to warm L2.
  __builtin_prefetch(logits + tid * 16, 0, 1);
  __builtin_prefetch(labels + tid * 16, 0, 1);

#if HAVE_TDM_BUILTIN
  if (wave == 0) {                         // TDM ignores EXEC; issue once per block
    tdm_load_1d(logits, s_logit4, n);
    tdm_load_1d(labels, s_label4, n);
    __builtin_amdgcn_s_wait_tensorcnt(0);  // TENSORcnt==0 before publishing
  }
  __syncthreads();
#else
  {
    // Async DMA fill: 16B per lane per issue, tracked by ASYNCcnt (ISA §10.1,
    // GLOBAL_LOAD_ASYNC_TO_LDS_B128). LDS dest addr = low 32b of generic ptr.
    const float4* gl4 = (const float4*)logits;
    const float4* gb4 = (const float4*)labels;
    const int n4 = n >> 2;
    for (int i = tid; i < n4; i += blockDim.x) {
      unsigned int l0 = (unsigned int)(uintptr_t)(s_logit4 + i);
      unsigned long long g0 = (unsigned long long)(uintptr_t)(gl4 + i);
      asm volatile("global_load_async_to_lds_b128 %0, %1, off"
                   :: "v"(l0), "v"(g0) : "memory");
      unsigned int l1 = (unsigned int)(uintptr_t)(s_label4 + i);
      unsigned long long g1 = (unsigned long long)(uintptr_t)(gb4 + i);
      asm volatile("global_load_async_to_lds_b128 %0, %1, off"
                   :: "v"(l1), "v"(g1) : "memory");
    }
    asm volatile("s_wait_asynccnt 0x0" ::: "memory");
    __syncthreads();
  }
#endif

  const int wavesPerBlock = blockDim.x >> 5;                 // 8
  const int waveG = blockIdx.x * wavesPerBlock + wave;       // 0..4095
  const int totalWaves = gridDim.x * wavesPerBlock;          // 4096
  const int tiles = n >> 4;                                  // 512

  // Row partition: wavesPerRow waves share one tile row (A operands hoisted).
  const int wavesPerRow = (totalWaves + tiles - 1) / tiles;  // 8
  const int rowStride = totalWaves / wavesPerRow;            // 512
  const int row0 = waveG / wavesPerRow;
  const int colOff = waveG % wavesPerRow;

  const int sub = lane & 15;   // matrix row/col index for this lane
  const bool lo = lane < 16;   // lanes 16..31 carry A cols K=2,3 -> forced to 0

  float acc = 0.0f;

  for (int ti = row0; ti < tiles; ti += rowStride) {
    // A (16x4): col0 = -ls_i / -lb_i (negation hoisted), col1 = ones,
    // cols 2,3 (lanes 16-31) = 0  => B rows 2,3 may hold arbitrary finite data.
    float lsi = s_logit[ti * 16 + sub];
    float lbi = s_label[ti * 16 + sub];
    float nlsA = lo ? -lsi : 0.0f;
    float nlbA = lo ? -lbi : 0.0f;
    float oneA = lo ? 1.0f : 0.0f;
    const v2f aL = {nlsA, oneA};
    const v2f aB = {nlbA, oneA};

    for (int tj = colOff; tj < tiles; tj += wavesPerRow) {
      // B (4x16): row0 = ones, row1 = ls_j / lb_j. Loads on ALL lanes —
      // lanes 16-31 feed B rows 2,3 which multiply A's zero columns.
      float lsB = s_logit[tj * 16 + sub];
      float lbB = s_label[tj * 16 + sub];
      const v2f bL = {1.0f, lsB};
      const v2f bB = {1.0f, lbB};
      const v8f c0 = {};

      // y[m,n]  = ls_j[n] - ls_i[m]  (= -diff);  yb = lb_j - lb_i (= -lbdiff)
      v8f y = __builtin_amdgcn_wmma_f32_16x16x4_f32(
          false, aL, false, bL, (short)0, c0, false, false);
      v8f yb = __builtin_amdgcn_wmma_f32_16x16x4_f32(
          false, aB, false, bB, (short)0, c0, false, false);

#pragma unroll
      for (int v = 0; v < 8; ++v) {
        float yy = y[v];
        // log_sigmoid(-yy) = -max(yy,0) - ln2*log2(1 + exp2(-|yy|*log2e))
        float e = fast_exp2(__builtin_fabsf(yy) * -1.44269504088896340736f);
        float l2 = fast_log2(1.0f + e);
        float ls = __builtin_fmaf(-0.69314718055994530942f, l2,
                                  -__builtin_fmaxf(yy, 0.0f));
        // keep iff lb_i - lb_j > TOL  <=>  yb < -TOL
        acc += (yb[v] < -TOLERANCE) ? ls : 0.0f;
      }
    }
  }

  // Wave32 reduction.
#pragma unroll
  for (int off = 16; off > 0; off >>= 1) acc += __shfl_xor(acc, off, 32);
  if (lane == 0) s_wavesum[wave] = acc;
  __syncthreads();
  if (tid == 0) {
    float s = 0.0f;
    for (int w = 0; w < wavesPerBlock; ++w) s += s_wavesum[w];
    partial[blockIdx.x] = s;
  }
}

__global__ __launch_bounds__(THREADS) void RankingLoss_reduce_kernel(
    const float* __restrict__ partial, int m, float* __restrict__ out) {
  __shared__ float s_wavesum[THREADS / 32];
  float a = 0.0f;
  for (int i = threadIdx.x; i < m; i += blockDim.x) a += partial[i];
#pragma unroll
  for (int off = 16; off > 0; off >>= 1) a += __shfl_xor(a, off, 32);
  const int lane = threadIdx.x & 31;
  const int wave = threadIdx.x >> 5;
  if (lane == 0) s_wavesum[wave] = a;
  __syncthreads();
  if (threadIdx.x == 0) {
    float t = 0.0f;
    for (int w = 0; w < (int)(blockDim.x >> 5); ++w) t += s_wavesum[w];
    out[0] = t;  // overwrite poison; deterministic (no atomics)
  }
}

extern "C" void kernel_launch(void* const* d_in, const int* in_sizes, int n_in,
                              void* d_out, int out_size, void* d_ws, size_t ws_size,
                              hipStream_t stream) {
  const float* logits = (const float*)d_in[0];
  const float* labels = (const float*)d_in[1];
  float* out = (float*)d_out;
  float* partial = (float*)d_ws;  // BLOCKS floats of scratch
  const int n = in_sizes[0];

  RankingLoss_pairs_kernel<<<BLOCKS, THREADS, 0, stream>>>(logits, labels, partial, n);
  RankingLoss_reduce_kernel<<<1, THREADS, 0, stream>>>(partial, BLOCKS, out);
}